// HierarchicalSoftmax_32126355374937
// MI455X (gfx1250) — compile-verified
//
#include <hip/hip_runtime.h>
#include <hip/hip_bf16.h>

typedef __attribute__((ext_vector_type(2))) float v2f;
typedef __attribute__((ext_vector_type(8))) float v8f;

#define B_SAMP 2048
#define D_DIM  512
#define NCLS   256
#define NOUT   256

// ---------------- prep kernels ----------------
__global__ void k_zero(float* __restrict__ zeros_row, int* __restrict__ counts) {
  int t = threadIdx.x;
  if (t < D_DIM) zeros_row[t] = 0.f;
  if (t < NCLS)  counts[t] = 0;
}

__global__ void k_hist(const int* __restrict__ target, int* __restrict__ tgt_class,
                       int* __restrict__ tgt_in, int* __restrict__ counts) {
  int i = blockIdx.x * blockDim.x + threadIdx.x;
  if (i >= B_SAMP) return;
  int t  = target[i];
  int tc = t >> 8;     // / NOUT
  int ti = t & 255;    // % NOUT
  tgt_class[i] = tc;
  tgt_in[i]    = ti;
  atomicAdd(&counts[tc], 1);
}

__global__ void k_scan(const int* __restrict__ counts, int* __restrict__ offsets,
                       int* __restrict__ cursor) {
  if (threadIdx.x == 0) {
    int run = 0;
    for (int c = 0; c < NCLS; ++c) { offsets[c] = run; cursor[c] = run; run += counts[c]; }
    offsets[NCLS] = run;
  }
}

__global__ void k_scatter(const int* __restrict__ tgt_class, int* __restrict__ cursor,
                          int* __restrict__ order) {
  int i = blockIdx.x * blockDim.x + threadIdx.x;
  if (i >= B_SAMP) return;
  int pos = atomicAdd(&cursor[tgt_class[i]], 1);
  order[pos] = i;
}

// ---------------- level-1 GEMM: logits1 = x @ W1 + b1 ----------------
// One wave (32 threads) computes one 16x16 f32 tile via V_WMMA_F32_16X16X4_F32.
__global__ __launch_bounds__(32) void k_l1_gemm(const float* __restrict__ x,
                                                const float* __restrict__ W1,
                                                const float* __restrict__ b1,
                                                float* __restrict__ logits1) {
  const int lane = threadIdx.x;
  const int half = lane >> 4;      // 0: K pair {0,1}, 1: K pair {2,3}
  const int l16  = lane & 15;
  const int mt   = blockIdx.x >> 4;   // 128 M-tiles
  const int nt   = blockIdx.x & 15;   // 16  N-tiles
  const int row  = mt * 16 + l16;
  const int col  = nt * 16 + l16;
  const float* xrow = x + (size_t)row * D_DIM;

  v8f acc = {};
#pragma unroll 4
  for (int k = 0; k < D_DIM; k += 4) {
    // A 16x4: lane holds row (l16), K = k + half*2 + {0,1}
    v2f a = *(const v2f*)(xrow + k + half * 2);
    // B 4x16: lane holds col, K = k + half*2 + {0,1}
    v2f bb;
    bb.x = W1[(size_t)(k + half * 2 + 0) * NCLS + col];
    bb.y = W1[(size_t)(k + half * 2 + 1) * NCLS + col];
    acc = __builtin_amdgcn_wmma_f32_16x16x4_f32(false, a, false, bb,
                                                (short)0, acc, false, false);
  }
  const float bias = b1[col];
#pragma unroll
  for (int r = 0; r < 8; ++r) {
    int m = mt * 16 + half * 8 + r;   // C layout: VGPR r -> M = r + half*8
    logits1[(size_t)m * NCLS + col] = acc[r] + bias;
  }
}

// ---------------- level-1 softmax, gather target-class prob ----------------
__global__ __launch_bounds__(NCLS) void k_l1_softmax(const float* __restrict__ logits1,
                                                     const int* __restrict__ tgt_class,
                                                     float* __restrict__ class_p) {
  __shared__ float red[NCLS];
  const int b = blockIdx.x, t = threadIdx.x;
  float v = logits1[(size_t)b * NCLS + t];
  red[t] = v; __syncthreads();
  for (int s = NCLS / 2; s > 0; s >>= 1) {
    if (t < s) red[t] = fmaxf(red[t], red[t + s]);
    __syncthreads();
  }
  float mx = red[0]; __syncthreads();
  float e = __expf(v - mx);
  red[t] = e; __syncthreads();
  for (int s = NCLS / 2; s > 0; s >>= 1) {
    if (t < s) red[t] += red[t + s];
    __syncthreads();
  }
  float sum = red[0];
  if (t == tgt_class[b]) class_p[b] = e / sum;
}

// ---------------- level-2 grouped GEMM: per class c, X_g @ W2[c] + b2[c] ----
// grid = NCLS * 16 waves; each wave owns (class, N-tile), loops over M-tiles.
__global__ __launch_bounds__(32) void k_l2_gemm(const float* __restrict__ x,
                                                const float* __restrict__ W2,
                                                const float* __restrict__ b2,
                                                const int* __restrict__ offsets,
                                                const int* __restrict__ order,
                                                const float* __restrict__ zeros_row,
                                                float* __restrict__ in_logits) {
  const int lane  = threadIdx.x;
  const int half  = lane >> 4;
  const int l16   = lane & 15;
  const int cls   = blockIdx.x >> 4;
  const int nt    = blockIdx.x & 15;
  const int start = offsets[cls];
  const int n_c   = offsets[cls + 1] - start;
  if (n_c == 0) return;                      // wave-uniform: EXEC stays all-ones

  const int col = nt * 16 + l16;
  const float* W2c  = W2 + (size_t)cls * D_DIM * NOUT;
  const float  bias = b2[(size_t)cls * NOUT + col];

  for (int mt = 0; mt * 16 < n_c; ++mt) {
    const int m = mt * 16 + l16;
    // pad short M-tiles with a zero row so WMMA runs with full EXEC
    const float* arow = (m < n_c) ? (x + (size_t)order[start + m] * D_DIM)
                                  : zeros_row;
    v8f acc = {};
#pragma unroll 4
    for (int k = 0; k < D_DIM; k += 4) {
      v2f a = *(const v2f*)(arow + k + half * 2);
      v2f bb;
      bb.x = W2c[(size_t)(k + half * 2 + 0) * NOUT + col];
      bb.y = W2c[(size_t)(k + half * 2 + 1) * NOUT + col];
      acc = __builtin_amdgcn_wmma_f32_16x16x4_f32(false, a, false, bb,
                                                  (short)0, acc, false, false);
    }
#pragma unroll
    for (int r = 0; r < 8; ++r) {
      int m2 = mt * 16 + half * 8 + r;
      if (m2 < n_c) {
        int samp = order[start + m2];
        in_logits[(size_t)samp * NOUT + col] = acc[r] + bias;
      }
    }
  }
}

// ---------------- level-2 softmax + final product ----------------
__global__ __launch_bounds__(NOUT) void k_l2_final(const float* __restrict__ in_logits,
                                                   const int* __restrict__ tgt_in,
                                                   const float* __restrict__ class_p,
                                                   float* __restrict__ out) {
  __shared__ float red[NOUT];
  const int b = blockIdx.x, t = threadIdx.x;
  float v = in_logits[(size_t)b * NOUT + t];
  red[t] = v; __syncthreads();
  for (int s = NOUT / 2; s > 0; s >>= 1) {
    if (t < s) red[t] = fmaxf(red[t], red[t + s]);
    __syncthreads();
  }
  float mx = red[0]; __syncthreads();
  float e = __expf(v - mx);
  red[t] = e; __syncthreads();
  for (int s = NOUT / 2; s > 0; s >>= 1) {
    if (t < s) red[t] += red[t + s];
    __syncthreads();
  }
  float sum = red[0];
  if (t == tgt_in[b]) out[b] = class_p[b] * (e / sum);
}

// ---------------- host side ----------------
extern "C" void kernel_launch(void* const* d_in, const int* in_sizes, int n_in,
                              void* d_out, int out_size, void* d_ws, size_t ws_size,
                              hipStream_t stream) {
  const float* x   = (const float*)d_in[0];
  const int*   tgt = (const int*)d_in[1];
  const float* W1  = (const float*)d_in[2];
  const float* b1  = (const float*)d_in[3];
  const float* W2  = (const float*)d_in[4];
  const float* b2  = (const float*)d_in[5];
  float* out = (float*)d_out;

  // workspace carve-up (all offsets 256B-aligned)
  char* ws = (char*)d_ws;
  size_t off = 0;
  float* logits1   = (float*)(ws + off); off += (size_t)B_SAMP * NCLS * 4;  // 2 MB
  float* in_logits = (float*)(ws + off); off += (size_t)B_SAMP * NOUT * 4;  // 2 MB
  float* class_p   = (float*)(ws + off); off += (size_t)B_SAMP * 4;
  float* zeros_row = (float*)(ws + off); off += (size_t)D_DIM * 4;
  int* tgt_class   = (int*)(ws + off);   off += (size_t)B_SAMP * 4;
  int* tgt_in      = (int*)(ws + off);   off += (size_t)B_SAMP * 4;
  int* order       = (int*)(ws + off);   off += (size_t)B_SAMP * 4;
  int* counts      = (int*)(ws + off);   off += (size_t)NCLS * 4;
  int* cursor      = (int*)(ws + off);   off += (size_t)NCLS * 4;
  int* offsets     = (int*)(ws + off);   off += (size_t)(NCLS + 1) * 4;
  (void)ws_size; (void)in_sizes; (void)n_in; (void)out_size;

  // 1) zero scratch + histogram targets by class
  k_zero<<<1, 512, 0, stream>>>(zeros_row, counts);
  k_hist<<<B_SAMP / 256, 256, 0, stream>>>(tgt, tgt_class, tgt_in, counts);
  k_scan<<<1, 32, 0, stream>>>(counts, offsets, cursor);
  k_scatter<<<B_SAMP / 256, 256, 0, stream>>>(tgt_class, cursor, order);

  // 2) level-1 GEMM (f32 WMMA) + softmax-gather
  k_l1_gemm<<<(B_SAMP / 16) * (NCLS / 16), 32, 0, stream>>>(x, W1, b1, logits1);
  k_l1_softmax<<<B_SAMP, NCLS, 0, stream>>>(logits1, tgt_class, class_p);

  // 3) level-2 grouped GEMM (f32 WMMA, W2 streamed exactly once) + final
  k_l2_gemm<<<NCLS * (NOUT / 16), 32, 0, stream>>>(x, W2, b2, offsets, order,
                                                   zeros_row, in_logits);
  k_l2_final<<<B_SAMP, NOUT, 0, stream>>>(in_logits, tgt_in, class_p, out);
}